// GCN_71966472011840
// MI455X (gfx1250) — compile-verified
//
#include <hip/hip_runtime.h>

// ---------------------------------------------------------------------------
// 2-layer GCN for MI455X (gfx1250, wave32).
//   h   = x @ W            -> fp32 WMMA (V_WMMA_F32_16X16X4_F32)
//   out = D^-1/2 (A+I) D^-1/2 h  -> edge-parallel gather + f32 global atomics
//   relu in place
// ---------------------------------------------------------------------------

typedef __attribute__((ext_vector_type(2))) float v2f;
typedef __attribute__((ext_vector_type(8))) float v8f;

#define GEMM_BLOCK 128  // 4 waves of 32

// H[M,N] = X[M,K] @ W[K,N].  K multiple of 4 (=256 here), N multiple of 64,
// M multiple of 16.  One wave computes a 16x64 output tile.
__global__ __launch_bounds__(GEMM_BLOCK)
void gcn_gemm_wmma(const float* __restrict__ X, const float* __restrict__ W,
                   float* __restrict__ H, int M, int K, int N) {
  // B panel (K x 64 cols) staged in LDS, pair-interleaved on K so each lane's
  // (k,k+1) B-fragment is one contiguous 64-bit LDS read.
  __shared__ float Bs[256 * 64];  // 64 KB of the 320 KB WGP LDS
  const int tid  = threadIdx.x;
  const int lane = tid & 31;
  const int wave = tid >> 5;
  const int n0   = blockIdx.y * 64;

  for (int idx = tid; idx < (K << 6); idx += GEMM_BLOCK) {
    int k = idx >> 6;   // row in K
    int c = idx & 63;   // col in panel
    Bs[((k >> 1) << 7) + (c << 1) + (k & 1)] = W[k * N + n0 + c];
  }
  __syncthreads();

  const int m0 = (blockIdx.x * 4 + wave) << 4;
  if (m0 >= M) return;

  // A fragment layout (16x4 f32): lanes 0-15 rows M=0..15 (K=k0,k0+1),
  // lanes 16-31 same rows (K=k0+2,k0+3).
  const int row  = m0 + (lane & 15);
  const int koff = (lane >> 4) << 1;        // 0 or 2
  const int col2 = (lane & 15) << 1;        // paired-B column offset

  v8f acc0 = {}, acc1 = {}, acc2 = {}, acc3 = {};
  const float* xrow = X + (long)row * K + koff;
  for (int k0 = 0; k0 < K; k0 += 4) {
    v2f a = *(const v2f*)(xrow + k0);                       // global_load_b64
    const float* bb = Bs + (((k0 + koff) >> 1) << 7) + col2;
    v2f b0 = *(const v2f*)(bb);                             // ds_load_b64 x4
    v2f b1 = *(const v2f*)(bb + 32);
    v2f b2 = *(const v2f*)(bb + 64);
    v2f b3 = *(const v2f*)(bb + 96);
    acc0 = __builtin_amdgcn_wmma_f32_16x16x4_f32(false, a, false, b0, (short)0, acc0, false, false);
    acc1 = __builtin_amdgcn_wmma_f32_16x16x4_f32(false, a, false, b1, (short)0, acc1, false, false);
    acc2 = __builtin_amdgcn_wmma_f32_16x16x4_f32(false, a, false, b2, (short)0, acc2, false, false);
    acc3 = __builtin_amdgcn_wmma_f32_16x16x4_f32(false, a, false, b3, (short)0, acc3, false, false);
  }

  // C/D layout: VGPR i -> row m0 + (lane>=16 ? 8 : 0) + i, col = lane&15.
  const long rbase = m0 + ((lane >> 4) << 3);
  float* o = H + rbase * N + n0 + (lane & 15);
#pragma unroll
  for (int i = 0; i < 8; ++i) {
    long r = (long)i * N;
    o[r +  0] = acc0[i];
    o[r + 16] = acc1[i];
    o[r + 32] = acc2[i];
    o[r + 48] = acc3[i];
  }
}

// ---- degree / normalization (shared by both layers) -----------------------

__global__ __launch_bounds__(256)
void gcn_deg_init(float* __restrict__ deg, int n) {
  int i = blockIdx.x * 256 + threadIdx.x;
  if (i < n) deg[i] = 1.0f;  // self-loop
}

__global__ __launch_bounds__(256)
void gcn_deg_count(const int* __restrict__ dst, float* __restrict__ deg, int E) {
  int e = blockIdx.x * 256 + threadIdx.x;
  if (e < E) atomicAdd(&deg[dst[e]], 1.0f);
}

__global__ __launch_bounds__(256)
void gcn_deg_rsqrt(float* __restrict__ deg, int n) {
  int i = blockIdx.x * 256 + threadIdx.x;
  if (i < n) deg[i] = rsqrtf(deg[i]);  // deg >= 1 always
}

// ---- aggregation ----------------------------------------------------------

// Self-loop term, also initializes the output: out[i,:] = h[i,:] * dinv[i]^2
// blockDim.x == F (256 or 64), one block per node.
__global__ void gcn_agg_self(const float* __restrict__ h,
                             const float* __restrict__ dinv,
                             float* __restrict__ out) {
  int node = blockIdx.x;
  int F = blockDim.x;
  float d = dinv[node];
  long i = (long)node * F + threadIdx.x;
  out[i] = h[i] * d * d;
}

// Edge-parallel scatter: F/4 lanes per edge, float4 gather + 4 f32 atomics.
// lpeShift = log2(F/4): 6 for F=256, 4 for F=64.
__global__ __launch_bounds__(256)
void gcn_agg_edges(const float* __restrict__ h, const int* __restrict__ src,
                   const int* __restrict__ dst, const float* __restrict__ dinv,
                   float* __restrict__ out, int E, int F, int lpeShift) {
  int lane = threadIdx.x & ((1 << lpeShift) - 1);
  long e = (long)blockIdx.x * (256 >> lpeShift) + (threadIdx.x >> lpeShift);
  if (e >= E) return;
  int s = src[e];
  int d = dst[e];
  float norm = dinv[s] * dinv[d];
  const float4 v = ((const float4*)(h + (long)s * F))[lane];  // global_load_b128
  float* ob = out + (long)d * F + (lane << 2);
  atomicAdd(ob + 0, v.x * norm);   // global_atomic_add_f32 (L2-resident)
  atomicAdd(ob + 1, v.y * norm);
  atomicAdd(ob + 2, v.z * norm);
  atomicAdd(ob + 3, v.w * norm);
}

__global__ __launch_bounds__(256)
void gcn_relu(float* __restrict__ p, long total) {
  long i = blockIdx.x * 256L + threadIdx.x;
  if (i < total) {
    float v = p[i];
    p[i] = v > 0.0f ? v : 0.0f;
  }
}

// ---------------------------------------------------------------------------

extern "C" void kernel_launch(void* const* d_in, const int* in_sizes, int n_in,
                              void* d_out, int out_size, void* d_ws, size_t ws_size,
                              hipStream_t stream) {
  const float* x  = (const float*)d_in[0];   // [n, 256]
  const int*   ei = (const int*)d_in[1];     // [2, E] (harness integer ABI)
  const float* W1 = (const float*)d_in[2];   // [256, 256]
  const float* W3 = (const float*)d_in[3];   // [256, 64]

  const int n = in_sizes[0] / 256;
  const int E = in_sizes[1] / 2;
  const int* src = ei;
  const int* dst = ei + E;

  // workspace: dinv | bufA (n x 256) | bufB (n x 256)
  char* ws = (char*)d_ws;
  float* dinv = (float*)ws;
  size_t off = (((size_t)n * 4 + 255) / 256) * 256;
  float* bufA = (float*)(ws + off);
  float* bufB = bufA + (size_t)n * 256;
  float* out  = (float*)d_out;

  // ---- degree normalization (same for both layers) ----
  gcn_deg_init <<<(n + 255) / 256, 256, 0, stream>>>(dinv, n);
  gcn_deg_count<<<(E + 255) / 256, 256, 0, stream>>>(dst, dinv, E);
  gcn_deg_rsqrt<<<(n + 255) / 256, 256, 0, stream>>>(dinv, n);

  const int mblocks = ((n + 15) / 16 + 3) / 4;

  // ---- layer 1: h1 = relu(agg(x @ W1)), F = 256 ----
  gcn_gemm_wmma<<<dim3(mblocks, 256 / 64), GEMM_BLOCK, 0, stream>>>(
      x, W1, bufA, n, 256, 256);
  gcn_agg_self <<<n, 256, 0, stream>>>(bufA, dinv, bufB);
  gcn_agg_edges<<<(E + 3) / 4, 256, 0, stream>>>(bufA, src, dst, dinv, bufB, E, 256, 6);
  gcn_relu     <<<(int)(((long)n * 256 + 255) / 256), 256, 0, stream>>>(bufB, (long)n * 256);

  // ---- layer 2: out = relu(agg(h1 @ W3)), F = 64 ----
  gcn_gemm_wmma<<<dim3(mblocks, 1), GEMM_BLOCK, 0, stream>>>(
      bufB, W3, bufA, n, 256, 64);
  gcn_agg_self <<<n, 64, 0, stream>>>(bufA, dinv, out);
  gcn_agg_edges<<<(E + 15) / 16, 256, 0, stream>>>(bufA, src, dst, dinv, out, E, 64, 4);
  gcn_relu     <<<(int)(((long)n * 64 + 255) / 256), 256, 0, stream>>>(out, (long)n * 64);
}